// prospectdcore_81097572483403
// MI455X (gfx1250) — compile-verified
//
#include <hip/hip_runtime.h>

typedef __attribute__((ext_vector_type(2))) float v2f;
typedef __attribute__((ext_vector_type(8))) float v8f;

#define NUM_LEAVES 8192
#define N_LAMBDA   2101
#define TILES_W    ((N_LAMBDA + 15) / 16)   // 132
#define TILES_M    (NUM_LEAVES / 16)        // 512

// ---- fast math on CDNA5 TRANS units (~1 ulp; rcp refined to ~0.5 ulp) ----
__device__ __forceinline__ float frcp(float x) {
    float r = __builtin_amdgcn_rcpf(x);
    float e = __builtin_fmaf(-x, r, 1.0f);   // 1 - x*r
    return __builtin_fmaf(r, e, r);          // one Newton-Raphson step
}
__device__ __forceinline__ float fdiv(float a, float b) { return a * frcp(b); }
__device__ __forceinline__ float fexp(float x) { return __builtin_amdgcn_exp2f(x * 1.4426950408889634f); }
__device__ __forceinline__ float flog(float x) { return __builtin_amdgcn_logf(x) * 0.6931471805599453f; }
__device__ __forceinline__ float fsqrt(float x) { return __builtin_amdgcn_sqrtf(x); }

// Transmissivity of a dielectric plane (Stern / Allen), per-wavelength only.
__device__ __forceinline__ float getav(float sa2, bool is90, float nr) {
    float n2  = nr * nr;
    float npx = n2 + 1.0f;
    float nm  = n2 - 1.0f;
    float a   = (nr + 1.0f) * (nr + 1.0f) * 0.5f;
    float k   = -(nm * nm) * 0.25f;
    float b2  = sa2 - npx * 0.5f;
    float b1  = is90 ? 0.0f : fsqrt(b2 * b2 + k);
    float b   = b1 - b2;
    float k2  = k * k;
    float inva = frcp(a);
    float invb = frcp(b);
    float ts  = k2 * (invb * invb * invb) * (1.0f / 6.0f) + k * invb - b * 0.5f
              - (k2 * (inva * inva * inva) * (1.0f / 6.0f) + k * inva - a * 0.5f);
    float nm2 = nm * nm;
    float invnpx  = frcp(npx);
    float invnpx2 = invnpx * invnpx;
    float tp1 = -2.0f * n2 * (b - a) * invnpx2;
    float tp2 = -2.0f * n2 * npx * flog(b * inva) * frcp(nm2);
    float tp3 = n2 * (invb - inva) * 0.5f;
    float n22 = n2 * n2;
    float invnpx3 = invnpx2 * invnpx;
    float npax_a = 2.0f * npx * a - nm2;
    float npx_b  = 2.0f * npx * b - nm2;
    float inv_npax_a = frcp(npax_a);
    float inv_npx_b  = frcp(npx_b);
    float tp4 = 16.0f * n22 * (n22 + 1.0f) * flog(npx_b * inv_npax_a) * (invnpx3 * frcp(nm2));
    float tp5 = 16.0f * n22 * n2 * (inv_npx_b - inv_npax_a) * invnpx3;
    float tp  = tp1 + tp2 + tp3 + tp4 + tp5;
    return (ts + tp) * frcp(2.0f * sa2);
}

__global__ __launch_bounds__(256) void prospectd_wmma_kernel(
    const float* __restrict__ Np,   const float* __restrict__ cab,
    const float* __restrict__ car,  const float* __restrict__ water,
    const float* __restrict__ lma,  const float* __restrict__ cant,
    const float* __restrict__ nr,   const float* __restrict__ kab,
    const float* __restrict__ kcar, const float* __restrict__ kant,
    const float* __restrict__ kw,   const float* __restrict__ km,
    float* __restrict__ out)
{
    const int lane   = threadIdx.x & 31;
    const int waveId = blockIdx.x * 8 + (threadIdx.x >> 5);
    const int tile_m = waveId / TILES_W;
    const int tile_n = waveId - tile_m * TILES_W;
    if (tile_m >= TILES_M) return;

    const int lbase = tile_m << 4;          // first leaf of tile
    const int nbase = tile_n << 4;          // first wavelength of tile
    const int half  = lane >> 4;            // 0: K=0,1 / rows 0..7 ; 1: K=2,3 / rows 8..15
    const int mn    = lane & 15;
    const int wl    = nbase + mn;
    const int wlc   = (wl < N_LAMBDA) ? wl : (N_LAMBDA - 1);

    // ---- A matrix: traits scaled by 1/N. 16x4 f32 layout:
    // lanes 0-15 hold row M=lane, K={0,1}; lanes 16-31 hold row M=lane-16, K={2,3}.
    const int leafA = lbase + mn;
    const float Ninv = frcp(Np[leafA]);
    float ax, ay, a2x;
    if (half == 0) { ax = cab[leafA]  * Ninv; ay = car[leafA]   * Ninv; a2x = lma[leafA] * Ninv; }
    else           { ax = cant[leafA] * Ninv; ay = water[leafA] * Ninv; a2x = 0.0f; }

    // ---- B matrix: spectra rows K0=kab,K1=kcar,K2=kant,K3=kw ; second WMMA K0=km.
    float bx, by, b2x;
    if (half == 0) { bx = kab[wlc];  by = kcar[wlc]; b2x = km[wlc]; }
    else           { bx = kant[wlc]; by = kw[wlc];   b2x = 0.0f; }

    v2f A1 = {ax,  ay};
    v2f B1 = {bx,  by};
    v2f A2 = {a2x, 0.0f};
    v2f B2 = {b2x, 0.0f};
    v8f acc = {};
    // kall = (traits/N) * spectra : rank-5 GEMM as two f32 WMMA ops
    acc = __builtin_amdgcn_wmma_f32_16x16x4_f32(false, A1, false, B1, (short)0, acc, false, false);
    acc = __builtin_amdgcn_wmma_f32_16x16x4_f32(false, A2, false, B2, (short)0, acc, false, false);

    // ---- Per-wavelength surface optics (hoisted out of the 8-element loop).
    const float nrv   = nr[wlc];
    const float talf  = getav(0.41317591f, false, nrv);   // sin^2(40 deg)
    const float t12   = getav(1.0f, true, nrv);           // alpha = 90 deg
    const float ralf  = 1.0f - talf;
    const float r12   = 1.0f - t12;
    const float t21   = t12 * frcp(nrv * nrv);
    const float r21   = 1.0f - t21;
    const float r21sq = r21 * r21;

    // N for the 8 leaf rows this lane's accumulators cover (M = moff + v).
    const int moff = half << 3;
    const float4 nA = *reinterpret_cast<const float4*>(Np + lbase + moff);
    const float4 nB = *reinterpret_cast<const float4*>(Np + lbase + moff + 4);
    const float Nv[8] = {nA.x, nA.y, nA.z, nA.w, nB.x, nB.y, nB.z, nB.w};

    float reflv[8], tranv[8];
#pragma unroll
    for (int v = 0; v < 8; ++v) {
        float k  = fmaxf(acc[v], 1e-4f);
        float ek = fexp(-k);

        // E1(k): A&S 5.1.53 (k<=1) / 5.1.56 (k>1)
        float pa = 0.00107857f;
        pa = pa * k - 0.00976004f;
        pa = pa * k + 0.05519968f;
        pa = pa * k - 0.24991055f;
        pa = pa * k + 0.99999193f;
        pa = pa * k - 0.57721566f;
        float e1s = pa - flog(k);
        float pn = k + 8.5733287401f;
        pn = pn * k + 18.0590169730f;
        pn = pn * k + 8.6347608925f;
        pn = pn * k + 0.2677737343f;
        float pd = k + 9.5733223454f;
        pd = pd * k + 25.6329561486f;
        pd = pd * k + 21.0996530827f;
        pd = pd * k + 3.9584969228f;
        float e1b = (ek * pn) * frcp(k * pd);
        float e1  = (k <= 1.0f) ? e1s : e1b;
        float tau = (1.0f - k) * ek + k * k * e1;

        // one-layer R/T: share 1/denom across Ta and t
        float inv_denom = frcp(1.0f - r21sq * tau * tau);
        float c   = tau * t21 * inv_denom;
        float Ta  = talf * c;
        float t   = t12  * c;
        float r21tau = r21 * tau;
        float Ra  = ralf + r21tau * Ta;
        float r   = r12  + r21tau * t;

        float rpt = r + t;
        float rmt = r - t;
        bool  mask = (rpt >= 1.0f);
        float Dsq = (1.0f + rpt) * (1.0f + rmt) * (1.0f - rmt) * (1.0f - rpt);
        float D   = fsqrt(mask ? 1.0f : Dsq);
        float rqmtq = r * r - t * t;
        // a = (1+rq+D)/(2r), b = (1-rq+D)/(2t): share 1/(2rt)
        float inv2rt = frcp(2.0f * r * t);
        float a_ = (1.0f + rqmtq + D) * t * inv2rt;
        float b_ = (1.0f - rqmtq + D) * r * inv2rt;
        float b_safe = mask ? 1.0f : b_;
        float Nm1  = Nv[v] - 1.0f;
        float bnm1 = fexp(Nm1 * flog(b_safe));      // b_safe^(N-1)
        float bn2  = bnm1 * bnm1;
        float a2   = a_ * a_;
        float inv_den2 = frcp(a2 * bn2 - 1.0f);     // shared across Rsub/Tsub
        float Rsub = a_ * (bn2 - 1.0f) * inv_den2;
        float Tsub = bnm1 * (a2 - 1.0f) * inv_den2;
        float Tsub_alt = t * frcp(t + (1.0f - t) * Nm1);
        Tsub = mask ? Tsub_alt : Tsub;
        Rsub = mask ? (1.0f - Tsub_alt) : Rsub;
        float inv_den3 = frcp(1.0f - Rsub * r);     // shared across refl/tran
        tranv[v] = Ta * Tsub * inv_den3;
        reflv[v] = Ra + Ta * Rsub * t * inv_den3;
    }

    if (wl < N_LAMBDA) {
        const size_t LW = (size_t)NUM_LEAVES * (size_t)N_LAMBDA;
        float* __restrict__ outR = out + (size_t)(lbase + moff) * (size_t)N_LAMBDA + (size_t)wl;
#pragma unroll
        for (int v = 0; v < 8; ++v) {
            outR[(size_t)v * N_LAMBDA]      = reflv[v];   // refl
            outR[LW + (size_t)v * N_LAMBDA] = tranv[v];   // tran
        }
    }
}

extern "C" void kernel_launch(void* const* d_in, const int* in_sizes, int n_in,
                              void* d_out, int out_size, void* d_ws, size_t ws_size,
                              hipStream_t stream) {
    const float* Np    = (const float*)d_in[0];
    const float* cab   = (const float*)d_in[1];
    const float* car   = (const float*)d_in[2];
    const float* water = (const float*)d_in[3];
    const float* lma   = (const float*)d_in[4];
    const float* cant  = (const float*)d_in[5];
    const float* nr    = (const float*)d_in[6];
    const float* kab   = (const float*)d_in[7];
    const float* kcar  = (const float*)d_in[8];
    const float* kant  = (const float*)d_in[9];
    const float* kw    = (const float*)d_in[10];
    const float* km    = (const float*)d_in[11];
    float* out = (float*)d_out;

    const int totalWaves = TILES_M * TILES_W;        // 512 * 132 = 67584
    const int blocks = (totalWaves + 7) / 8;         // 8 waves (256 threads) per block
    prospectd_wmma_kernel<<<blocks, 256, 0, stream>>>(
        Np, cab, car, water, lma, cant, nr, kab, kcar, kant, kw, km, out);
}